// torchTripLoss_11991548690923
// MI455X (gfx1250) — compile-verified
//
#include <hip/hip_runtime.h>
#include <hip/hip_bf16.h>

// ---------------------------------------------------------------------------
// Triplet-style margin loss on MI455X (gfx1250).
// Memory-bound: x = 128MB read twice (centroid pass + WMMA dist pass) ->
// ~11.5us HBM floor at 23.3 TB/s. dist matrix = 16MB, L2-resident for the
// selection passes. GEMM is 1.07 GFLOP fp32 -> full-precision
// V_WMMA_F32_16X16X4_F32 (wave32), two independent accumulator chains,
// fed by double-buffered GLOBAL_LOAD_ASYNC_TO_LDS_B128 staging (ASYNCcnt).
// ---------------------------------------------------------------------------

#define C_CLS 16
#define DIMS  128
#define KSEL  102
#define EPS   1e-6f
#define DWAVES 4          // waves per block in dist kernel
#define TILE_BYTES (16 * DIMS * 4)   // 8KB: 16 columns x 128 dims fp32

typedef __attribute__((ext_vector_type(2))) float v2f;
typedef __attribute__((ext_vector_type(8))) float v8f;

// ---------------- kernel 0: zero the small accumulators ---------------------
__global__ void zero_accum_kernel(float* __restrict__ csum, unsigned* __restrict__ ccnt) {
    const int t = threadIdx.x;
    for (int i = t; i < C_CLS * DIMS; i += 256) csum[i] = 0.0f;
    if (t < C_CLS) ccnt[t] = 0u;
}

// ---------------- kernel 1: per-class sums + counts (one x pass) ------------
__global__ void center_accum_kernel(const float* __restrict__ x,
                                    const int* __restrict__ y,
                                    float* __restrict__ csum,
                                    unsigned* __restrict__ ccnt,
                                    int N) {
    __shared__ float    lsum[C_CLS * DIMS];
    __shared__ unsigned lcnt[C_CLS];
    const int t = threadIdx.x;
    for (int i = t; i < C_CLS * DIMS; i += blockDim.x) lsum[i] = 0.0f;
    if (t < C_CLS) lcnt[t] = 0u;
    __syncthreads();

    const int d   = t & (DIMS - 1);   // dim 0..127
    const int sub = t >> 7;           // 0/1: two rows processed per step
    for (int r = blockIdx.x * 2 + sub; r < N; r += gridDim.x * 2) {
        const int cls = y[3 * r + 2];                       // y2
        atomicAdd(&lsum[cls * DIMS + d], x[(size_t)r * DIMS + d]);
        if (d == 0) atomicAdd(&lcnt[cls], 1u);
    }
    __syncthreads();
    for (int i = t; i < C_CLS * DIMS; i += blockDim.x) atomicAdd(&csum[i], lsum[i]);
    if (t < C_CLS) atomicAdd(&ccnt[t], lcnt[t]);
}

// ---------------- kernel 2: finalize centers (in place) + adjC --------------
__global__ void finalize_centers_kernel(float* __restrict__ centers,  // csum in/out
                                        const unsigned* __restrict__ ccnt,
                                        float* __restrict__ adjC) {
    const int t = threadIdx.x;
    for (int i = t; i < C_CLS * DIMS; i += 256) {
        const int c = i >> 7;
        centers[i] = centers[i] / (float)ccnt[c];
    }
    __syncthreads();
    if (t < C_CLS) {
        float s = 0.0f, s2 = 0.0f;
        for (int d = 0; d < DIMS; ++d) {
            const float v = centers[t * DIMS + d];
            s += v; s2 += v * v;
        }
        // per-class constant of d2: c2 + 2*eps*sc + D*eps^2
        adjC[t] = s2 + 2.0f * EPS * s + (float)DIMS * EPS * EPS;
    }
}

// ---- async staging: contiguous 8KB x-tile -> LDS, 16 x b128 per wave -------
__device__ inline void async_copy_tile(unsigned ldsBase, const char* gBase, int lane) {
    const unsigned l0 = ldsBase + (unsigned)lane * 16u;
    const char*    g0 = gBase + lane * 16;
#pragma unroll
    for (int i = 0; i < 16; ++i) {
        asm volatile("global_load_async_to_lds_b128 %0, %1, off"
                     :: "v"(l0 + (unsigned)(i * 512)),
                        "v"(g0 + (size_t)(i * 512))
                     : "memory");
    }
}

// ---------------- kernel 3: WMMA fp32 dist matrix (one x pass) --------------
// One wave32 handles a strided set of 16-column tiles. Each tile (8KB of x,
// contiguous) is DMA'd to LDS with async b128 copies, double-buffered so the
// next tile's DMA overlaps the current tile's WMMA work. K=128 is split into
// two independent 16-deep V_WMMA_F32_16X16X4_F32 accumulation chains for
// matrix-pipe ILP. A-fragments (centers) live in registers for the whole wave.
__global__ void __launch_bounds__(DWAVES * 32)
dist_kernel(const float* __restrict__ x,
            const float* __restrict__ centers,
            const float* __restrict__ adjC,
            float* __restrict__ dist,
            int N) {
    __shared__ float xtile[DWAVES * 2 * 16 * DIMS];   // 4 waves x 2 bufs x 8KB = 64KB

    const int lane  = threadIdx.x & 31;
    const int wave  = threadIdx.x >> 5;
    const int wgid  = blockIdx.x * DWAVES + wave;     // global wave id
    const int nWav  = gridDim.x * DWAVES;
    const int m     = lane & 15;   // A row / B column handled by this lane
    const int khalf = lane >> 4;   // 0: K={0,1} of each 4-block, 1: K={2,3}

    float* buf0 = &xtile[wave * 2 * 16 * DIMS];
    float* buf1 = buf0 + 16 * DIMS;
    const unsigned lds0 = (unsigned)(uintptr_t)(void*)buf0;
    const unsigned lds1 = (unsigned)(uintptr_t)(void*)buf1;

    // preload all A fragments (centers row m) into 64 VGPRs, reused per tile
    v2f afr[32];
    {
        const float* __restrict__ crow = centers + m * DIMS;
#pragma unroll
        for (int kb = 0; kb < 32; ++kb) {
            const int kd = kb * 4 + khalf * 2;
            afr[kb].x = crow[kd];
            afr[kb].y = crow[kd + 1];
        }
    }
    float adj[8];
#pragma unroll
    for (int v = 0; v < 8; ++v) adj[v] = adjC[v + khalf * 8];

    const int nTiles = N / 16;
    const int tilesPerWave = nTiles / nWav;
    int tile = wgid;

    // prime the pipeline: tile 0 -> buf0
    async_copy_tile(lds0, (const char*)x + (size_t)tile * TILE_BYTES, lane);

    for (int it = 0; it < tilesPerWave; ++it) {
        const int nextTile = tile + nWav;
        const float* curBuf = (it & 1) ? buf1 : buf0;

        if (it + 1 < tilesPerWave) {
            const unsigned nxtLds = (it & 1) ? lds0 : lds1;
            async_copy_tile(nxtLds, (const char*)x + (size_t)nextTile * TILE_BYTES, lane);
            // 16 issues outstanding for next tile; current tile's 16 are done
            asm volatile("s_wait_asynccnt 16" ::: "memory");
        } else {
            asm volatile("s_wait_asynccnt 0" ::: "memory");
        }

        v8f acc0 = {}, acc1 = {};                       // two independent chains
        float ps1x = 0.0f, ps1y = 0.0f, ps2x = 0.0f, ps2y = 0.0f;
        const float* __restrict__ brow = curBuf + m * DIMS;
#pragma unroll
        for (int kb = 0; kb < 32; kb += 2) {
            const int kd0 = kb * 4 + khalf * 2;         // B 4x16 f32 layout
            const int kd1 = kd0 + 4;
            const v2f b0 = *(const v2f*)&brow[kd0];     // ds_load (2addr-paired)
            const v2f b1 = *(const v2f*)&brow[kd1];
            ps1x += b0.x; ps1y += b0.y;
            ps2x += b0.x * b0.x; ps2y += b0.y * b0.y;
            ps1x += b1.x; ps1y += b1.y;
            ps2x += b1.x * b1.x; ps2y += b1.y * b1.y;
            acc0 = __builtin_amdgcn_wmma_f32_16x16x4_f32(
                false, afr[kb],     false, b0, (short)0, acc0, false, false);
            acc1 = __builtin_amdgcn_wmma_f32_16x16x4_f32(
                false, afr[kb + 1], false, b1, (short)0, acc1, false, false);
        }
        // lanes L and L^16 together cover all 128 dims of column m
        const float ps1 = ps1x + ps1y;
        const float ps2 = ps2x + ps2y;
        const float q1 = ps1 + __shfl_xor(ps1, 16, 32);
        const float q2 = ps2 + __shfl_xor(ps2, 16, 32);
        const float colterm = q2 - 2.0f * EPS * q1;     // x2 - 2*eps*sx

        const int n = tile * 16 + m;
#pragma unroll
        for (int v = 0; v < 8; ++v) {
            const int cls = v + khalf * 8;              // C/D VGPR layout
            const float dot = acc0[v] + acc1[v];
            const float d2 = adj[v] + colterm - 2.0f * dot;
            dist[(size_t)cls * N + n] = sqrtf(fmaxf(d2, 0.0f));
        }
        tile = nextTile;
    }
}

// ---------------- kernel 4: exact top-K via radix-select + bitonic ----------
__device__ inline unsigned make_key(float dval, int lbl, int c, int pol) {
    const bool valid = pol ? (lbl != c) : (lbl == c);
    if (!valid) return 0xFFFFFFFFu;
    const unsigned u = __float_as_uint(dval) | 0x80000000u;   // dval >= 0
    return pol ? u : ~u;   // pos: descending (complement), neg: ascending
}

__global__ void select_topk_kernel(const float* __restrict__ dist,
                                   const int* __restrict__ y,
                                   float* __restrict__ topOut,  // [2][16][KSEL]
                                   int N) {
    __shared__ unsigned hist[256];
    __shared__ unsigned lbuf[128];
    __shared__ unsigned sprefix, srank, scnt;

    const int c   = blockIdx.x & 15;
    const int pol = blockIdx.x >> 4;   // 0 = positives (y2==c), 1 = negatives (y0!=c)
    const int t   = threadIdx.x;
    const float* __restrict__ drow = dist + (size_t)c * N;
    const int ycol = pol ? 0 : 2;

    if (t == 0) { sprefix = 0u; srank = KSEL - 1; }

    // 4-round MSB radix select of the K-th smallest key
    for (int r = 0; r < 4; ++r) {
        const int shift = 24 - 8 * r;
        hist[t] = 0u;
        __syncthreads();
        const unsigned prefix = sprefix;
        const unsigned hmask  = (r == 0) ? 0u : (0xFFFFFFFFu << (shift + 8));
        for (int n = t; n < N; n += 256) {
            const unsigned key = make_key(drow[n], y[3 * n + ycol], c, pol);
            if ((key & hmask) == (prefix & hmask))
                atomicAdd(&hist[(key >> shift) & 255u], 1u);
        }
        __syncthreads();
        if (t == 0) {
            unsigned rank = srank, run = 0u, g = 0u;
            for (; g < 256u; ++g) {
                const unsigned h = hist[g];
                if (run + h > rank) { srank = rank - run; break; }
                run += h;
            }
            sprefix = prefix | (g << shift);
        }
        __syncthreads();
    }

    const unsigned T = sprefix;        // exact K-th smallest key
    if (t == 0) scnt = 0u;
    __syncthreads();
    // gather strictly-smaller keys (count < K by construction)
    for (int n = t; n < N; n += 256) {
        const unsigned key = make_key(drow[n], y[3 * n + ycol], c, pol);
        if (key < T) {
            const unsigned p = atomicAdd(&scnt, 1u);
            if (p < 128u) lbuf[p] = key;
        }
    }
    __syncthreads();
    const unsigned base = scnt;
    if (t >= (int)base && t < 128) lbuf[t] = (t < KSEL) ? T : 0xFFFFFFFFu;
    __syncthreads();

    // bitonic sort 128 keys ascending
    for (int k = 2; k <= 128; k <<= 1) {
        for (int j = k >> 1; j > 0; j >>= 1) {
            if (t < 128) {
                const int ixj = t ^ j;
                if (ixj > t) {
                    const bool up = ((t & k) == 0);
                    const unsigned a = lbuf[t], b = lbuf[ixj];
                    if ((a > b) == up) { lbuf[t] = b; lbuf[ixj] = a; }
                }
            }
            __syncthreads();
        }
    }

    if (t < KSEL) {
        const unsigned key = lbuf[t];
        const unsigned u   = pol ? key : ~key;
        topOut[(size_t)pol * (C_CLS * KSEL) + c * KSEL + t] =
            __uint_as_float(u & 0x7FFFFFFFu);
    }
}

// ---------------- kernel 5: final margin-loss reduction ---------------------
__global__ void loss_kernel(const float* __restrict__ topOut, float* __restrict__ out) {
    __shared__ float red[256];
    const int t = threadIdx.x;
    const float* tp = topOut;                    // top_pos (descending)
    const float* tn = topOut + C_CLS * KSEL;     // top_neg (ascending)
    float s = 0.0f;
    for (int e = t; e < C_CLS * KSEL; e += 256)
        s += fmaxf(tp[e] - tn[e] + 1.0f, 0.0f);
    red[t] = s;
    __syncthreads();
    for (int w = 128; w > 0; w >>= 1) {
        if (t < w) red[t] += red[t + w];
        __syncthreads();
    }
    if (t == 0) out[0] = red[0] / (float)(C_CLS * KSEL);
}

// ---------------------------------------------------------------------------
extern "C" void kernel_launch(void* const* d_in, const int* in_sizes, int n_in,
                              void* d_out, int out_size, void* d_ws, size_t ws_size,
                              hipStream_t stream) {
    const float* x = (const float*)d_in[0];
    const int*   y = (const int*)d_in[1];
    float* out = (float*)d_out;
    const int N = in_sizes[0] / DIMS;          // 262144

    char* ws = (char*)d_ws;
    float*    csum   = (float*)(ws);                 // 16*128 f32, becomes centers
    unsigned* ccnt   = (unsigned*)(ws + 8192);       // 16 u32
    float*    adjC   = (float*)(ws + 8256);          // 16 f32
    float*    topOut = (float*)(ws + 8320);          // 2*16*102 f32
    float*    dist   = (float*)(ws + 32768);         // 16*N f32  (~16 MB)

    zero_accum_kernel<<<1, 256, 0, stream>>>(csum, ccnt);
    center_accum_kernel<<<512, 256, 0, stream>>>(x, y, csum, ccnt, N);
    finalize_centers_kernel<<<1, 256, 0, stream>>>(csum, ccnt, adjC);

    // 512 blocks x 4 waves = 2048 waves; 16384 tiles -> 8 tiles/wave,
    // double-buffered async-to-LDS pipeline per wave.
    dist_kernel<<<512, DWAVES * 32, 0, stream>>>(x, csum, adjC, dist, N);

    select_topk_kernel<<<32, 256, 0, stream>>>(dist, y, topOut, N);
    loss_kernel<<<1, 256, 0, stream>>>(topOut, out);
}